// MinibatchDiscrimination_18416819765386
// MI455X (gfx1250) — compile-verified
//
#include <hip/hip_runtime.h>
#include <stdint.h>

typedef __attribute__((ext_vector_type(2))) float v2f;
typedef __attribute__((ext_vector_type(8))) float v8f;

#define IN_FEAT 512
#define OUT_FEAT 768
#define BFEAT 256          // discrimination features
#define IDIM 16            // intermediate dim
#define NBATCH 256         // batch
#define COLS (BFEAT * IDIM)  // 4096 flattened GEMM columns

// ---------------------------------------------------------------------------
// Kernel 1: M(256x4096) = x(256x512) @ W(512x4096), W = T flattened.
// One wave -> 16x64 output strip = 4 accumulators, V_WMMA_F32_16X16X4_F32,
// K walked in steps of 4 (128 iterations). A fragment reused across the 4
// column tiles; the 4 WMMAs per step are independent (no D->A/B hazard).
// ---------------------------------------------------------------------------
__global__ __launch_bounds__(256) void gemm_wmma_f32(
    const float* __restrict__ x, const float* __restrict__ T,
    float* __restrict__ M) {
  const int wave = threadIdx.x >> 5;
  const int lane = threadIdx.x & 31;
  const int g = blockIdx.x * 8 + wave;   // 0..1023 waves total
  const int rowTile = g >> 6;            // 0..15  (16 row tiles of 16)
  const int colGrp  = g & 63;            // 0..63  (64 col groups of 64)
  const int n0 = rowTile * 16;
  const int c0 = colGrp * 64;
  const int half = lane >> 4;            // 0: K pair {0,1} / rows 0-7
  const int l15  = lane & 15;            // row (A) / col (B,D) within tile

  // A 16x4 f32 layout: lanes 0-15 hold row=lane, K={k,k+1}; lanes 16-31 row=lane-16, K={k+2,k+3}
  const float* aPtr = x + (size_t)(n0 + l15) * IN_FEAT + half * 2;
  // B 4x16 layout (row striped across lanes): VGPR0 = W[k + 2*half][col], VGPR1 = W[k + 2*half + 1][col]
  const float* bPtr = T + (size_t)(half * 2) * COLS + c0 + l15;

  v8f acc0 = {}, acc1 = {}, acc2 = {}, acc3 = {};
  for (int k = 0; k < IN_FEAT; k += 4) {
    v2f a;
    a.x = aPtr[k];
    a.y = aPtr[k + 1];
    const float* bp = bPtr + (size_t)k * COLS;
    v2f b0, b1, b2, b3;
    b0.x = bp[0];   b0.y = bp[COLS + 0];
    b1.x = bp[16];  b1.y = bp[COLS + 16];
    b2.x = bp[32];  b2.y = bp[COLS + 32];
    b3.x = bp[48];  b3.y = bp[COLS + 48];
    acc0 = __builtin_amdgcn_wmma_f32_16x16x4_f32(false, a, false, b0, (short)0, acc0, false, false);
    acc1 = __builtin_amdgcn_wmma_f32_16x16x4_f32(false, a, false, b1, (short)0, acc1, false, false);
    acc2 = __builtin_amdgcn_wmma_f32_16x16x4_f32(false, a, false, b2, (short)0, acc2, false, false);
    acc3 = __builtin_amdgcn_wmma_f32_16x16x4_f32(false, a, false, b3, (short)0, acc3, false, false);
  }

  // D layout: VGPR v holds row v (lanes 0-15) and row v+8 (lanes 16-31), col = l15.
  float* dBase = M + (size_t)(n0 + half * 8) * COLS + c0 + l15;
#pragma unroll
  for (int v = 0; v < 8; ++v) {
    dBase[(size_t)v * COLS + 0]  = acc0[v];
    dBase[(size_t)v * COLS + 16] = acc1[v];
    dBase[(size_t)v * COLS + 32] = acc2[v];
    dBase[(size_t)v * COLS + 48] = acc3[v];
  }
}

// ---------------------------------------------------------------------------
// Kernel 2: one block per feature b. Stage M[:, b, :] (256x16 f32 = 16 KB)
// into LDS via CDNA5 async global->LDS copies, then each thread owns row n,
// loops j over 256 broadcast LDS rows accumulating exp(-L1).
// ---------------------------------------------------------------------------
__global__ __launch_bounds__(256) void pairwise_l1exp(
    const float* __restrict__ M, float* __restrict__ out) {
  __shared__ __align__(16) float tile[NBATCH * IDIM];  // 16 KB
  const int b = blockIdx.x;
  const int n = threadIdx.x;

  // Each thread async-copies its own 64-byte row M[n, b, 0..15] into LDS.
  unsigned ldsAddr = (unsigned)(uintptr_t)(&tile[n * IDIM]);  // low 32 bits = LDS offset
  unsigned memOff  = (unsigned)(((size_t)n * COLS + (size_t)b * IDIM) * sizeof(float));
  asm volatile(
      "global_load_async_to_lds_b128 %0, %1, %2\n\t"
      "global_load_async_to_lds_b128 %0, %1, %2 offset:16\n\t"
      "global_load_async_to_lds_b128 %0, %1, %2 offset:32\n\t"
      "global_load_async_to_lds_b128 %0, %1, %2 offset:48"
      :
      : "v"(ldsAddr), "v"(memOff), "s"(M)
      : "memory");
  asm volatile("s_wait_asynccnt 0" ::: "memory");
  __syncthreads();

  float m[IDIM];
#pragma unroll
  for (int i = 0; i < IDIM; ++i) m[i] = tile[n * IDIM + i];

  float sum = 0.0f;
  for (int j = 0; j < NBATCH; ++j) {
    const float* r = &tile[j * IDIM];  // broadcast read: all lanes same addr
    float l1 = 0.0f;
#pragma unroll
    for (int i = 0; i < IDIM; ++i) l1 += fabsf(m[i] - r[i]);
    sum += __expf(-l1);
  }
  out[(size_t)n * OUT_FEAT + IN_FEAT + b] = sum - 1.0f;  // -1: remove self pair
}

// ---------------------------------------------------------------------------
// Kernel 3: out[:, 0:512] = x  (float4 copies, rows stay 16B aligned)
// ---------------------------------------------------------------------------
__global__ __launch_bounds__(256) void copy_x_rows(
    const float4* __restrict__ x4, float* __restrict__ out) {
  int idx = blockIdx.x * 256 + threadIdx.x;  // 0 .. 256*128-1
  int n  = idx >> 7;                         // row
  int f4 = idx & 127;                        // float4 slot within row
  float4 v = x4[idx];
  *(float4*)(out + (size_t)n * OUT_FEAT + f4 * 4) = v;
}

extern "C" void kernel_launch(void* const* d_in, const int* in_sizes, int n_in,
                              void* d_out, int out_size, void* d_ws, size_t ws_size,
                              hipStream_t stream) {
  const float* x = (const float*)d_in[0];   // (256, 512) f32
  const float* T = (const float*)d_in[1];   // (512, 256, 16) f32 == 512x4096
  float* out = (float*)d_out;               // (256, 768) f32
  float* M = (float*)d_ws;                  // scratch: 256*4096 f32 = 4 MiB

  gemm_wmma_f32<<<128, 256, 0, stream>>>(x, T, M);      // 1024 waves, 8/block
  copy_x_rows<<<128, 256, 0, stream>>>((const float4*)x, out);
  pairwise_l1exp<<<256, 256, 0, stream>>>(M, out);      // one block per feature b
}